// InfluenceEncoder_76656576299328
// MI455X (gfx1250) — compile-verified
//
#include <hip/hip_runtime.h>

typedef float v2f __attribute__((ext_vector_type(2)));
typedef float v8f __attribute__((ext_vector_type(8)));

#define IN_DIM  128
#define HID_DIM 128

// ---------------------------------------------------------------------------
// Kernel 1: zero the edge counter (workspace is poisoned, not zeroed, by harness)
// ---------------------------------------------------------------------------
__global__ void ie_init(int* __restrict__ counter) { *counter = 0; }

// ---------------------------------------------------------------------------
// Kernel 2: scan dest[] for edges pointing at the ego node; collect (src, w).
// int4 loads: 4 dests per thread -> 12.8 MB streamed at near-HBM rate.
// ---------------------------------------------------------------------------
__global__ void ie_scan(const int* __restrict__ ei, const float* __restrict__ ew,
                        const int* __restrict__ ego_p, int E,
                        int* __restrict__ counter, int* __restrict__ src_list,
                        float* __restrict__ w_list, int cap) {
  int t = blockIdx.x * blockDim.x + threadIdx.x;
  int base = t * 4;
  if (base >= E) return;
  const int ego = *ego_p;
  const int* __restrict__ dest = ei + E;   // edge_index row 1
  if (base + 3 < E) {
    int4 d = *(const int4*)(dest + base);
    int dv[4] = {d.x, d.y, d.z, d.w};
#pragma unroll
    for (int i = 0; i < 4; ++i) {
      if (dv[i] == ego) {
        int pos = atomicAdd(counter, 1);
        if (pos < cap) { src_list[pos] = ei[base + i]; w_list[pos] = ew[base + i]; }
      }
    }
  } else {
    for (int e = base; e < E; ++e) {
      if (dest[e] == ego) {
        int pos = atomicAdd(counter, 1);
        if (pos < cap) { src_list[pos] = ei[e]; w_list[pos] = ew[e]; }
      }
    }
  }
}

// ---------------------------------------------------------------------------
// Kernel 3: one block, 8 wave32s. Wave w owns output columns [16w, 16w+16).
// Stage A: [n x 128] gathered rows @ W1 via V_WMMA_F32_16X16X4_F32,
//          + bias, ReLU, edge-weighted reduction over rows -> agg[128] (LDS).
// Stage B: relu(agg) @ W2 + b2 via the same f32 WMMA (A = 1 live row).
// EXEC stays all-ones around every WMMA (selects compile to v_cndmask).
// ---------------------------------------------------------------------------
__global__ void __launch_bounds__(256) ie_gemm(
    const float* __restrict__ nf, const float* __restrict__ W1,
    const float* __restrict__ b1, const float* __restrict__ W2,
    const float* __restrict__ b2, const int* __restrict__ counter,
    const int* __restrict__ src_list, const float* __restrict__ w_list,
    float* __restrict__ out, int cap) {
  __shared__ float sacc[HID_DIM];

  const int tid  = threadIdx.x;
  const int wave = tid >> 5;       // 0..7 -> column tile
  const int lane = tid & 31;
  const int half = lane >> 4;      // 0: K 0/1 half, 1: K 2/3 half
  const int l15  = lane & 15;      // A: row M ; B/C: col N (within tile)
  const int col  = wave * 16 + l15;

  int n = *counter;
  if (n > cap) n = cap;
  const int ntiles = (n + 15) >> 4;

  const float bn = b1[col];
  float colacc = 0.f;

  for (int t = 0; t < ntiles; ++t) {
    const int row = t * 16 + l15;              // row < cap (cap multiple of 16)
    int sr = src_list[row];                    // in-bounds load, then select
    int srcn = (row < n) ? sr : 0;             // avoid OOB nf gather on padding
    const float* __restrict__ arow = nf + (size_t)srcn * IN_DIM;

    v8f c = {0.f, 0.f, 0.f, 0.f, 0.f, 0.f, 0.f, 0.f};
#pragma unroll
    for (int kk = 0; kk < IN_DIM / 4; ++kk) {
      const int kb = kk * 4 + half * 2;
      v2f a, b;
      a.x = arow[kb + 0];                      // A: M=l15, K=kb..kb+1
      a.y = arow[kb + 1];
      b.x = W1[(kb + 0) * HID_DIM + col];      // B: K=kb..kb+1, N=col
      b.y = W1[(kb + 1) * HID_DIM + col];
      c = __builtin_amdgcn_wmma_f32_16x16x4_f32(false, a, false, b,
                                                (short)0, c, false, false);
    }
    // C[v]: M = v + 8*half, N = col. ReLU(+bias), weight by edge, reduce rows.
#pragma unroll
    for (int v = 0; v < 8; ++v) {
      const int grow = t * 16 + v + half * 8;
      float wv = w_list[grow];                 // in-bounds (grow < cap)
      wv = (grow < n) ? wv : 0.f;              // padding rows contribute 0
      float h = c[v] + bn;
      h = h > 0.f ? h : 0.f;
      colacc += wv * h;
    }
  }

  // Each column is split across lane L (M rows 0..7) and L+16 (M rows 8..15).
  colacc += __shfl_xor(colacc, 16, 32);
  if (lane < 16) sacc[col] = colacc;
  __syncthreads();

  // Stage B: out = relu(sacc) @ W2 + b2. A-matrix: row 0 = relu(sacc), rest 0.
  v8f c2 = {0.f, 0.f, 0.f, 0.f, 0.f, 0.f, 0.f, 0.f};
#pragma unroll
  for (int kk = 0; kk < HID_DIM / 4; ++kk) {
    const int kb = kk * 4 + half * 2;
    float x0 = sacc[kb + 0]; x0 = x0 > 0.f ? x0 : 0.f;
    float x1 = sacc[kb + 1]; x1 = x1 > 0.f ? x1 : 0.f;
    v2f a, b;
    a.x = (l15 == 0) ? x0 : 0.f;               // only A-row M=0 is live
    a.y = (l15 == 0) ? x1 : 0.f;
    b.x = W2[(kb + 0) * HID_DIM + col];
    b.y = W2[(kb + 1) * HID_DIM + col];
    c2 = __builtin_amdgcn_wmma_f32_16x16x4_f32(false, a, false, b,
                                               (short)0, c2, false, false);
  }
  // Result row M=0 lives in C VGPR0 on lanes 0..15.
  if (lane < 16) out[col] = c2[0] + b2[col];
}

// ---------------------------------------------------------------------------
// Launch. Inputs: 0 nf[N,128] f32, 1 edge_index[2,E] i32, 2 ew[E] f32,
// 3 ego i32(scalar), 4 W1[128,128], 5 b1[128], 6 W2[128,128], 7 b2[128].
// Output: 128 f32. Workspace: [counter @0][src_list @64][w_list @64+4*cap].
// ---------------------------------------------------------------------------
extern "C" void kernel_launch(void* const* d_in, const int* in_sizes, int n_in,
                              void* d_out, int out_size, void* d_ws, size_t ws_size,
                              hipStream_t stream) {
  const float* nf  = (const float*)d_in[0];
  const int*   ei  = (const int*)d_in[1];
  const float* ew  = (const float*)d_in[2];
  const int*   ego = (const int*)d_in[3];
  const float* W1  = (const float*)d_in[4];
  const float* b1  = (const float*)d_in[5];
  const float* W2  = (const float*)d_in[6];
  const float* b2  = (const float*)d_in[7];

  const int E = in_sizes[1] / 2;

  int cap = 2048;  // mean matches ~= E/N = 32; 2048 is absurdly safe headroom
  if (ws_size < 64 + (size_t)cap * 8) {
    cap = (ws_size > 64) ? (int)((ws_size - 64) / 8) : 0;
  }
  cap &= ~15;      // multiple of 16 so padded-tile loads stay inside the lists

  int*   counter  = (int*)d_ws;
  int*   src_list = (int*)((char*)d_ws + 64);
  float* w_list   = (float*)((char*)d_ws + 64 + (size_t)cap * 4);

  ie_init<<<1, 1, 0, stream>>>(counter);

  const int quads  = (E + 3) / 4;
  const int blocks = (quads + 255) / 256;
  ie_scan<<<blocks, 256, 0, stream>>>(ei, ew, ego, E, counter, src_list, w_list, cap);

  ie_gemm<<<1, 256, 0, stream>>>(nf, W1, b1, W2, b2, counter, src_list, w_list,
                                 (float*)d_out, cap);
}